// AttentionalPlanarRemapping_75634374082662
// MI455X (gfx1250) — compile-verified
//
#include <hip/hip_runtime.h>
#include <math.h>

// Problem constants (from reference): N=32, C=64, H=W=128, E=512
#define NB     32
#define CDIM   64
#define HW     16384      // 128*128
#define EDIM   512
#define CC     4096       // C*C

typedef __attribute__((ext_vector_type(2))) float v2f;
typedef __attribute__((ext_vector_type(8))) float v8f;

// ---------------------------------------------------------------------------
// Kernel 1: logits = atts @ W^T + b ; softmax over 4096 ; reshape [64,64] ;
// softmax over last 64 ; write a[n,64,64] to workspace.
// One block per n. 256 threads = 8 waves (wave32).
// ---------------------------------------------------------------------------
__global__ __launch_bounds__(256) void attn_softmax_kernel(
    const float* __restrict__ atts, const float* __restrict__ W,
    const float* __restrict__ bias, float* __restrict__ a_out)
{
    __shared__ float s_att[EDIM];   // 2 KB
    __shared__ float s_log[CC];     // 16 KB
    __shared__ float s_red[8];

    const int n    = blockIdx.x;
    const int t    = threadIdx.x;
    const int wave = t >> 5;
    const int lane = t & 31;

    // stage atts[n,:] into LDS
    for (int i = t; i < EDIM; i += 256) s_att[i] = atts[(size_t)n * EDIM + i];
    __syncthreads();

    // logits: one wave per output j; lanes split K (coalesced 128B reads of W row)
    for (int j = wave; j < CC; j += 8) {
        const float* wrow = W + (size_t)j * EDIM;
        float acc = 0.f;
        #pragma unroll 4
        for (int e = lane; e < EDIM; e += 32)
            acc = fmaf(s_att[e], wrow[e], acc);
        #pragma unroll
        for (int off = 16; off > 0; off >>= 1)
            acc += __shfl_xor(acc, off, 32);
        if (lane == 0) s_log[j] = acc + bias[j];
    }
    __syncthreads();

    // ---- softmax over all 4096 ----
    float mx = -INFINITY;
    for (int i = t; i < CC; i += 256) mx = fmaxf(mx, s_log[i]);
    #pragma unroll
    for (int off = 16; off > 0; off >>= 1) mx = fmaxf(mx, __shfl_xor(mx, off, 32));
    if (lane == 0) s_red[wave] = mx;
    __syncthreads();
    float bmx = s_red[0];
    #pragma unroll
    for (int w = 1; w < 8; ++w) bmx = fmaxf(bmx, s_red[w]);
    __syncthreads();                 // done reading s_red before reuse

    float sm = 0.f;
    for (int i = t; i < CC; i += 256) {
        float ev = expf(s_log[i] - bmx);
        s_log[i] = ev;
        sm += ev;
    }
    #pragma unroll
    for (int off = 16; off > 0; off >>= 1) sm += __shfl_xor(sm, off, 32);
    if (lane == 0) s_red[wave] = sm;
    __syncthreads();
    float bsum = 0.f;
    #pragma unroll
    for (int w = 0; w < 8; ++w) bsum += s_red[w];
    const float inv = 1.0f / bsum;
    for (int i = t; i < CC; i += 256) s_log[i] *= inv;
    __syncthreads();

    // ---- second softmax over rows of 64 (one thread per row) ----
    if (t < CDIM) {
        const float* row = &s_log[t * CDIM];
        float rmx = -INFINITY;
        for (int k = 0; k < CDIM; ++k) rmx = fmaxf(rmx, row[k]);
        float rs = 0.f;
        for (int k = 0; k < CDIM; ++k) rs += expf(row[k] - rmx);
        const float rinv = 1.0f / rs;
        float* orow = a_out + (size_t)n * CC + (size_t)t * CDIM;
        for (int k = 0; k < CDIM; ++k) orow[k] = expf(row[k] - rmx) * rinv;
    }
}

// ---------------------------------------------------------------------------
// Kernel 2: out[n] = a[n] (64x64) @ images[n] (64x16384) via
// V_WMMA_F32_16X16X4_F32. Block = 256 threads = 8 waves; each wave owns a
// 64(M) x 16(N) output strip (4 stacked 16x16 tiles sharing one B fragment),
// so each image element is loaded from HBM exactly once. a[n] staged in LDS.
//
// fp32 WMMA fragment mapping (ISA 7.12.2), lane l, vgpr v:
//   A(16x4):  m = l&15,          k = k0 + v + 2*(l>=16)
//   B(4x16):  ncol = l&15,       k = k0 + v + 2*(l>=16)
//   C/D vgpr r: M = r + 8*(l>=16), N = l&15
// ---------------------------------------------------------------------------
__global__ __launch_bounds__(256) void remap_wmma_kernel(
    const float* __restrict__ a, const float* __restrict__ img,
    float* __restrict__ out)
{
    __shared__ float s_a[CDIM * CDIM];   // 16 KB

    const int n = blockIdx.y;
    const int t = threadIdx.x;

    // stage a[n] (64x64) into LDS, row-major
    const float* an = a + (size_t)n * CC;
    #pragma unroll
    for (int i = t; i < CC; i += 256) s_a[i] = an[i];
    __syncthreads();

    const int wave  = t >> 5;
    const int lane  = t & 31;
    const int half  = lane >> 4;     // 0: lanes 0-15, 1: lanes 16-31
    const int col16 = lane & 15;
    const int hw0   = blockIdx.x * 128 + wave * 16 + col16;   // this lane's N column

    const float* imgn = img + (size_t)n * CDIM * HW;

    v8f c0 = {}, c1 = {}, c2 = {}, c3 = {};

    #pragma unroll 4
    for (int k0 = 0; k0 < CDIM; k0 += 4) {
        const int kA = k0 + 2 * half;          // k for vgpr v: kA + v

        // B fragment: images rows kA, kA+1 at column hw0
        v2f bf;
        bf.x = imgn[(size_t)(kA + 0) * HW + hw0];
        bf.y = imgn[(size_t)(kA + 1) * HW + hw0];

        // A fragments for the 4 stacked M-tiles (8-byte aligned LDS loads)
        v2f a0 = *(const v2f*)&s_a[( 0 + col16) * CDIM + kA];
        v2f a1 = *(const v2f*)&s_a[(16 + col16) * CDIM + kA];
        v2f a2 = *(const v2f*)&s_a[(32 + col16) * CDIM + kA];
        v2f a3 = *(const v2f*)&s_a[(48 + col16) * CDIM + kA];

        c0 = __builtin_amdgcn_wmma_f32_16x16x4_f32(false, a0, false, bf, (short)0, c0, false, false);
        c1 = __builtin_amdgcn_wmma_f32_16x16x4_f32(false, a1, false, bf, (short)0, c1, false, false);
        c2 = __builtin_amdgcn_wmma_f32_16x16x4_f32(false, a2, false, bf, (short)0, c2, false, false);
        c3 = __builtin_amdgcn_wmma_f32_16x16x4_f32(false, a3, false, bf, (short)0, c3, false, false);
    }

    // store: vgpr r -> output row (mtile*16 + r + 8*half), column hw0
    float* outn = out + (size_t)n * CDIM * HW;
    #pragma unroll
    for (int r = 0; r < 8; ++r) {
        outn[(size_t)( 0 + r + 8 * half) * HW + hw0] = c0[r];
        outn[(size_t)(16 + r + 8 * half) * HW + hw0] = c1[r];
        outn[(size_t)(32 + r + 8 * half) * HW + hw0] = c2[r];
        outn[(size_t)(48 + r + 8 * half) * HW + hw0] = c3[r];
    }
}

// ---------------------------------------------------------------------------
extern "C" void kernel_launch(void* const* d_in, const int* in_sizes, int n_in,
                              void* d_out, int out_size, void* d_ws, size_t ws_size,
                              hipStream_t stream) {
    const float* images = (const float*)d_in[0];   // [32,64,128,128]
    const float* atts   = (const float*)d_in[1];   // [32,512]
    const float* W      = (const float*)d_in[2];   // [4096,512]
    const float* bias   = (const float*)d_in[3];   // [4096]
    float*       out    = (float*)d_out;           // [32,64,128,128]
    float*       a_ws   = (float*)d_ws;            // 32*4096 floats = 512 KB

    attn_softmax_kernel<<<NB, 256, 0, stream>>>(atts, W, bias, a_ws);

    dim3 grid(HW / 128, NB);                       // 128 x 32 blocks
    remap_wmma_kernel<<<grid, 256, 0, stream>>>(a_ws, images, out);
}